// NaNHandlingLayer_45861660786920
// MI455X (gfx1250) — compile-verified
//
#include <hip/hip_runtime.h>

typedef float v2f __attribute__((ext_vector_type(2)));
typedef float v8f __attribute__((ext_vector_type(8)));
typedef unsigned int u32x4 __attribute__((ext_vector_type(4)));
typedef int i32x4 __attribute__((ext_vector_type(4)));
typedef int i32x8 __attribute__((ext_vector_type(8)));

#define D_DIM 512
#define K_DIM 64

#if defined(__AMDGCN__) && __has_builtin(__builtin_amdgcn_tensor_load_to_lds)
#define USE_TDM 1
#else
#define USE_TDM 0
#endif

// ---------------------------------------------------------------------------
// Kernel 1: GT[n][d] = sum_{j<10} F[d][j] * W[j][n]   (transposed for WMMA-B)
// ---------------------------------------------------------------------------
__global__ __launch_bounds__(256) void prep_gt_kernel(
    const float* __restrict__ F,    // [D,10]
    const float* __restrict__ W,    // [11,K]
    float* __restrict__ GT)         // [K,D]
{
  int idx = blockIdx.x * blockDim.x + threadIdx.x;
  if (idx >= K_DIM * D_DIM) return;
  int n = idx >> 9;            // output-feature index 0..63
  int d = idx & (D_DIM - 1);   // input-feature index 0..511
  float acc = 0.f;
#pragma unroll
  for (int j = 0; j < 10; ++j)
    acc = fmaf(F[d * 10 + j], W[j * K_DIM + n], acc);
  GT[n * D_DIM + d] = acc;
}

// ---------------------------------------------------------------------------
// Kernel 2: out = relu( (mask*obs) @ G  +  s1 ⊗ W10  +  s0 ⊗ bw )
// One wave computes a 32(M) x 64(N) tile with V_WMMA_F32_16X16X4_F32.
// Block = 4 waves = 128 rows; GT + b staged in LDS (130 KB of 320 KB/WGP).
// GT staging uses the Tensor Data Mover (tensor_load_to_lds) when available.
// ---------------------------------------------------------------------------
__global__ __launch_bounds__(128) void fused_masked_encode_kernel(
    const float* __restrict__ obs,   // [B,D]
    const float* __restrict__ mask,  // [B,D]
    const float* __restrict__ bvec,  // [D]
    const float* __restrict__ W,     // [11,K] (row 10 used)
    const float* __restrict__ bw,    // [K]
    const float* __restrict__ GT,    // [K,D]
    float* __restrict__ out)         // [B,K]
{
  __shared__ float ldsGT[K_DIM * D_DIM];  // 128 KB
  __shared__ float ldsB[D_DIM];           // 2 KB

  const int tid  = threadIdx.x;  // 0..127
  const int wave = tid >> 5;
  const int lane = tid & 31;

#if USE_TDM
  // --- TDM prologue: one wave issues a single 2-D descriptor that DMAs the
  // --- whole 64x512 f32 GT tile into LDS; tracked by TENSORcnt.
  if (wave == 0) {
    const uint64_t ga     = (uint64_t)(uintptr_t)GT;
    const uint32_t ldsoff = (uint32_t)(uintptr_t)&ldsGT[0];  // flat->LDS offset
    u32x4 g0;
    g0[0] = 1u;                                   // count=1, user D#
    g0[1] = ldsoff;                               // lds_addr
    g0[2] = (uint32_t)ga;                         // global_addr[31:0]
    g0[3] = (uint32_t)(ga >> 32) | (2u << 30);    // global_addr[56:32] | type=2
    i32x8 g1;
    g1[0] = (int)(2u << 16);                      // data_size = 4 bytes
    g1[1] = (int)(512u << 16);                    // tensor_dim0 = 512
    g1[2] = (int)(64u << 16);                     // tensor_dim1 = 64
    g1[3] = (int)(512u << 16);                    // tile_dim0 = 512
    g1[4] = 64;                                   // tile_dim1 = 64
    g1[5] = 512;                                  // tensor_dim0_stride = 512
    g1[6] = 0;
    g1[7] = 0;
    i32x4 gz = {0, 0, 0, 0};                      // 2-D: groups 2/3 unused
#if __clang_major__ >= 23
    i32x8 gz8 = {0, 0, 0, 0, 0, 0, 0, 0};
    __builtin_amdgcn_tensor_load_to_lds(g0, g1, gz, gz, gz8, 0);
#else
    __builtin_amdgcn_tensor_load_to_lds(g0, g1, gz, gz, 0);
#endif
    __builtin_amdgcn_s_wait_tensorcnt((short)0);
  }
#else
  // Fallback: cooperative stage of GT into LDS (b128 loads)
  for (int i = tid * 4; i < K_DIM * D_DIM; i += 128 * 4)
    *(float4*)(&ldsGT[i]) = *(const float4*)(&GT[i]);
#endif
  {
    int i = tid * 4;
    if (i < D_DIM) *(float4*)(&ldsB[i]) = *(const float4*)(&bvec[i]);
  }
  __syncthreads();

  const int ls = lane & 15;   // sub-lane: row (A) / col (B/C)
  const int hi = lane >> 4;   // half-wave: selects K pair / row+8 in C
  const int m0 = blockIdx.x * 128 + wave * 32;   // first of 32 rows

  const float* obs0 = obs  + (size_t)(m0 + ls) * D_DIM;
  const float* msk0 = mask + (size_t)(m0 + ls) * D_DIM;
  const float* obs1 = obs0 + 16 * D_DIM;
  const float* msk1 = msk0 + 16 * D_DIM;

  v8f acc0[4] = {};   // rows m0..m0+15, N-tiles 0..3
  v8f acc1[4] = {};   // rows m0+16..m0+31
  float s0a = 0.f, s1a = 0.f, s0b = 0.f, s1b = 0.f;

#pragma unroll 2
  for (int d = 0; d < D_DIM; d += 4) {
    const int col = d + 2 * hi;              // lane's K pair
    v2f o0 = *(const v2f*)(obs0 + col);
    v2f q0 = *(const v2f*)(msk0 + col);
    v2f o1 = *(const v2f*)(obs1 + col);
    v2f q1 = *(const v2f*)(msk1 + col);
    v2f bb = *(const v2f*)(&ldsB[col]);

    v2f a0; a0.x = o0.x * q0.x; a0.y = o0.y * q0.y;
    v2f a1; a1.x = o1.x * q1.x; a1.y = o1.y * q1.y;

    s0a += q0.x + q0.y;
    s1a = fmaf(q0.x, bb.x, fmaf(q0.y, bb.y, s1a));
    s0b += q1.x + q1.y;
    s1b = fmaf(q1.x, bb.x, fmaf(q1.y, bb.y, s1b));

#pragma unroll
    for (int t = 0; t < 4; ++t) {
      v2f bm = *(const v2f*)(&ldsGT[(t * 16 + ls) * D_DIM + col]);
      acc0[t] = __builtin_amdgcn_wmma_f32_16x16x4_f32(
          false, a0, false, bm, (short)0, acc0[t], false, false);
      acc1[t] = __builtin_amdgcn_wmma_f32_16x16x4_f32(
          false, a1, false, bm, (short)0, acc1[t], false, false);
    }
  }

  // Combine the two half-wave column partitions: lane ls then holds full
  // row sums for row ls of its tile.
  s0a += __shfl_xor(s0a, 16, 32);
  s1a += __shfl_xor(s1a, 16, 32);
  s0b += __shfl_xor(s0b, 16, 32);
  s1b += __shfl_xor(s1b, 16, 32);

  float w10v[4], bwv[4];
#pragma unroll
  for (int t = 0; t < 4; ++t) {
    w10v[t] = W[10 * K_DIM + t * 16 + ls];
    bwv[t]  = bw[t * 16 + ls];
  }

#pragma unroll
  for (int r = 0; r < 8; ++r) {
    const int rowA = r + 8 * hi;                 // row within 16-row tile
    const float t0s0 = __shfl(s0a, rowA, 32);
    const float t0s1 = __shfl(s1a, rowA, 32);
    const float t1s0 = __shfl(s0b, rowA, 32);
    const float t1s1 = __shfl(s1b, rowA, 32);
#pragma unroll
    for (int t = 0; t < 4; ++t) {
      float v0 = acc0[t][r] + t0s1 * w10v[t] + t0s0 * bwv[t];
      float v1 = acc1[t][r] + t1s1 * w10v[t] + t1s0 * bwv[t];
      out[(size_t)(m0 + rowA) * K_DIM + t * 16 + ls]      = fmaxf(v0, 0.f);
      out[(size_t)(m0 + 16 + rowA) * K_DIM + t * 16 + ls] = fmaxf(v1, 0.f);
    }
  }
}

// ---------------------------------------------------------------------------
extern "C" void kernel_launch(void* const* d_in, const int* in_sizes, int n_in,
                              void* d_out, int out_size, void* d_ws, size_t ws_size,
                              hipStream_t stream) {
  const float* obs  = (const float*)d_in[0];  // [4096,512]
  const float* mask = (const float*)d_in[1];  // [4096,512]
  const float* F    = (const float*)d_in[2];  // [1,512,10]
  const float* bvec = (const float*)d_in[3];  // [1,512,1]
  const float* W    = (const float*)d_in[4];  // [11,64]
  const float* bw   = (const float*)d_in[5];  // [64]
  float* out = (float*)d_out;                 // [4096,64]
  float* GT  = (float*)d_ws;                  // [64,512] = 128 KB scratch

  prep_gt_kernel<<<(K_DIM * D_DIM + 255) / 256, 256, 0, stream>>>(F, W, GT);
  fused_masked_encode_kernel<<<32, 128, 0, stream>>>(obs, mask, bvec, W, bw,
                                                     GT, out);
}